// DemLocalization_13211319402664
// MI455X (gfx1250) — compile-verified
//
#include <hip/hip_runtime.h>
#include <hip/hip_bf16.h>

typedef float v2f __attribute__((ext_vector_type(2)));
typedef float v8f __attribute__((ext_vector_type(8)));

#define NN 256      // nodes
#define TT 30000    // time dim
#define HH 512      // hidden
#define LL 256      // latent

#define WMMA4(a, b, c) \
  __builtin_amdgcn_wmma_f32_16x16x4_f32(false, (a), false, (b), (short)0, (c), false, false)

// ---------------- graph adjacency: M = I + A, A[d][s] = #edges s->d ----------
__global__ void k_init_M(float* __restrict__ M) {
  int i = blockIdx.x, j = threadIdx.x;
  M[i * NN + j] = (i == j) ? 1.0f : 0.0f;
}

__global__ void k_build_adj(const int* __restrict__ idx, float* __restrict__ M, int E) {
  int e = blockIdx.x * blockDim.x + threadIdx.x;
  if (e < E) {
    int s = idx[e];
    int d = idx[E + e];
    atomicAdd(&M[d * NN + s], 1.0f);
  }
}

// ---------------- fp32 WMMA GEMM: C = A[MxK] @ B[KxN] (+bias, +relu) ---------
// One wave per block (blockDim=32) so all tile indices / loop bounds are scalar.
// Wave computes a 16x64 tile: 4 accumulators share each A fragment; K unrolled
// by 8 -> 8 V_WMMA_F32_16X16X4_F32 per loop body.
// Requires: M%16==0, N%64==0, K%(8*splitk)==0.
// splitk>1: slice ks writes partial plane C[ks*M*N + ...] (deterministic 2-pass).
__global__ __launch_bounds__(32) void k_gemm_wmma(
    const float* __restrict__ A, int lda,
    const float* __restrict__ B, int ldb,
    float* __restrict__ C, int ldc,
    const float* __restrict__ bias,
    int M, int N, int K, int relu, int splitk) {
  const int wid  = blockIdx.x;            // one wave per block -> scalar indices
  const int lane = threadIdx.x;           // 0..31
  const int nt    = N >> 6;               // 64-wide n tiles
  const int tiles = (M >> 4) * nt;
  const int tile  = wid % tiles;
  const int ks    = wid / tiles;
  const int Kc    = K / splitk;
  const int kbeg  = ks * Kc, kend = kbeg + Kc;
  const int tm = tile / nt, tn = tile % nt;
  const int m0 = tm << 4, n0 = tn << 6;
  const int half = lane >> 4;             // selects K pair {0,1} vs {2,3}; C rows +8
  const int l15  = lane & 15;

  const float* Arow = A + (size_t)(m0 + l15) * lda;          // A frag row
  const float* Bp   = B + n0 + l15 + (size_t)(2 * half) * ldb;

  v8f acc0 = {}, acc1 = {}, acc2 = {}, acc3 = {};

#define BFRAG(bp, j) ({ v2f _b; _b[0] = (bp)[16 * (j)]; _b[1] = (bp)[16 * (j) + ldb]; _b; })

  for (int k = kbeg; k < kend; k += 8) {
    // A fragments for k..k+3 and k+4..k+7 (8B-aligned: lda even, offsets even)
    v2f a0 = *(const v2f*)(Arow + k + 2 * half);
    v2f a1 = *(const v2f*)(Arow + k + 4 + 2 * half);
    const float* b0 = Bp + (size_t)k * ldb;        // rows k+2h, k+2h+1
    const float* b1 = b0 + (size_t)4 * ldb;        // rows k+4+2h, k+5+2h
    v2f p0 = BFRAG(b0, 0), p1 = BFRAG(b0, 1), p2 = BFRAG(b0, 2), p3 = BFRAG(b0, 3);
    v2f q0 = BFRAG(b1, 0), q1 = BFRAG(b1, 1), q2 = BFRAG(b1, 2), q3 = BFRAG(b1, 3);
    __builtin_prefetch(Arow + k + 256, 0, 0);      // stream A row ahead
    __builtin_prefetch(b0 + (size_t)16 * ldb, 0, 0); // stream B panel ahead
    acc0 = WMMA4(a0, p0, acc0);
    acc1 = WMMA4(a0, p1, acc1);
    acc2 = WMMA4(a0, p2, acc2);
    acc3 = WMMA4(a0, p3, acc3);
    acc0 = WMMA4(a1, q0, acc0);
    acc1 = WMMA4(a1, q1, acc1);
    acc2 = WMMA4(a1, q2, acc2);
    acc3 = WMMA4(a1, q3, acc3);
  }
#undef BFRAG

  // epilogue: row = m0 + r + 8*half, col = n0 + 16*j + l15
  if (splitk > 1) {
    float* Cp = C + (size_t)ks * M * N;
#pragma unroll
    for (int j = 0; j < 4; ++j) {
      v8f acc = (j == 0) ? acc0 : (j == 1) ? acc1 : (j == 2) ? acc2 : acc3;
      int col = n0 + 16 * j + l15;
#pragma unroll
      for (int r = 0; r < 8; ++r)
        Cp[(size_t)(m0 + r + 8 * half) * N + col] = acc[r];
    }
  } else {
#pragma unroll
    for (int j = 0; j < 4; ++j) {
      v8f acc = (j == 0) ? acc0 : (j == 1) ? acc1 : (j == 2) ? acc2 : acc3;
      int col = n0 + 16 * j + l15;
      float bv = bias ? bias[col] : 0.0f;
#pragma unroll
      for (int r = 0; r < 8; ++r) {
        float v = acc[r] + bv;
        if (relu) v = fmaxf(v, 0.0f);
        C[(size_t)(m0 + r + 8 * half) * ldc + col] = v;
      }
    }
  }
}

// deterministic fixed-order reduction of split-K partials
__global__ void k_reduce_slices(const float* __restrict__ part, float* __restrict__ dst,
                                int n, int slices) {
  int i = blockIdx.x * blockDim.x + threadIdx.x;
  if (i < n) {
    float s = 0.0f;
    for (int k = 0; k < slices; ++k) s += part[(size_t)k * n + i];
    dst[i] = s;
  }
}

// ---------------- scorer head: per-node dot(512) -> relu -> affine -> sigmoid
__global__ void k_region(const float* __restrict__ g4, const float* __restrict__ Wc2a,
                         const float* __restrict__ bc2a, const float* __restrict__ Wc2b,
                         const float* __restrict__ bc2b, float* __restrict__ out) {
  __shared__ float red[128];
  int i = blockIdx.x, t = threadIdx.x;
  float s = 0.0f;
  for (int j = t; j < HH; j += 128) s += g4[i * HH + j] * Wc2a[j];
  red[t] = s;
  __syncthreads();
  for (int w = 64; w > 0; w >>= 1) {
    if (t < w) red[t] += red[t + w];
    __syncthreads();
  }
  if (t == 0) {
    float v = fmaxf(red[0] + bc2a[0], 0.0f);
    v = v * Wc2b[0] + bc2b[0];
    out[1 + i] = 1.0f / (1.0f + __expf(-v));
  }
}

// ---------------- dementia head: dot(65536) -> sigmoid ----------------------
__global__ void k_dementia(const float* __restrict__ feat, const float* __restrict__ Wd,
                           const float* __restrict__ bd, float* __restrict__ out) {
  __shared__ float red[256];
  int t = threadIdx.x;
  float s = 0.0f;
  for (int j = t; j < NN * LL; j += 256) s += feat[j] * Wd[j];
  red[t] = s;
  __syncthreads();
  for (int w = 128; w > 0; w >>= 1) {
    if (t < w) red[t] += red[t + w];
    __syncthreads();
  }
  if (t == 0) out[0] = 1.0f / (1.0f + __expf(-(red[0] + bd[0])));
}

// ---------------- host-side launcher ----------------------------------------
static inline void gemm(const float* A, int lda, const float* B, int ldb, float* C,
                        int ldc, const float* bias, int M, int N, int K, int relu,
                        int splitk, hipStream_t s) {
  int blocks = (M / 16) * (N / 64) * splitk;   // one wave (32 threads) per block
  k_gemm_wmma<<<blocks, 32, 0, s>>>(A, lda, B, ldb, C, ldc, bias, M, N, K, relu, splitk);
}

extern "C" void kernel_launch(void* const* d_in, const int* in_sizes, int n_in,
                              void* d_out, int out_size, void* d_ws, size_t ws_size,
                              hipStream_t stream) {
  const float* X    = (const float*)d_in[0];
  const int*   idx  = (const int*)d_in[1];
  // d_in[2] = eeg_attr: ignored by GINConv in the reference
  const float* Ws1a = (const float*)d_in[3];
  const float* bs1a = (const float*)d_in[4];
  const float* Ws1b = (const float*)d_in[5];
  const float* bs1b = (const float*)d_in[6];
  const float* Ws2a = (const float*)d_in[7];
  const float* bs2a = (const float*)d_in[8];
  const float* Ws2b = (const float*)d_in[9];
  const float* bs2b = (const float*)d_in[10];
  const float* Wc1a = (const float*)d_in[11];
  const float* bc1a = (const float*)d_in[12];
  const float* Wc1b = (const float*)d_in[13];
  const float* bc1b = (const float*)d_in[14];
  const float* Wc2a = (const float*)d_in[15];
  const float* bc2a = (const float*)d_in[16];
  const float* Wc2b = (const float*)d_in[17];
  const float* bc2b = (const float*)d_in[18];
  const float* Wd   = (const float*)d_in[19];
  const float* bd   = (const float*)d_in[20];
  float* out = (float*)d_out;
  int E = in_sizes[1] / 2;

  const int SPLITK = 10;  // 30000/10 = 3000, multiple of 8; deterministic two-pass
  float* ws = (float*)d_ws;
  float* Mmat   = ws; ws += NN * NN;            // I + A
  float* xwpart = ws; ws += SPLITK * NN * HH;   // split-K partials of X @ Ws1a
  float* xw     = ws; ws += NN * HH;
  float* t1     = ws; ws += NN * HH;
  float* h      = ws; ws += NN * HH;
  float* g2     = ws; ws += NN * HH;
  float* t2     = ws; ws += NN * LL;
  float* feat   = ws; ws += NN * LL;
  float* g3     = ws; ws += NN * LL;
  float* t3     = ws; ws += NN * HH;
  float* s1     = ws; ws += NN * HH;
  float* g4     = ws; ws += NN * HH;

  // M = I + A from edge list
  k_init_M<<<NN, NN, 0, stream>>>(Mmat);
  k_build_adj<<<(E + 255) / 256, 256, 0, stream>>>(idx, Mmat, E);

  // ---- struct_featurizer conv1:  relu(M @ (X @ Ws1a) + b) @ Ws1b + b, relu --
  gemm(X, TT, Ws1a, HH, xwpart, HH, nullptr, NN, HH, TT, 0, SPLITK, stream);  // big GEMM
  k_reduce_slices<<<(NN * HH + 255) / 256, 256, 0, stream>>>(xwpart, xw, NN * HH, SPLITK);
  gemm(Mmat, NN, xw, HH, t1, HH, bs1a, NN, HH, NN, 1, 1, stream);   // relu(M@xw + bs1a)
  gemm(t1, HH, Ws1b, HH, h, HH, bs1b, NN, HH, HH, 1, 1, stream);    // h = relu(conv1 out)

  // ---- struct_featurizer conv2: feat = relu((M@h) @ Ws2a + b) @ Ws2b + b ----
  gemm(Mmat, NN, h, HH, g2, HH, nullptr, NN, HH, NN, 0, 1, stream);
  gemm(g2, HH, Ws2a, LL, t2, LL, bs2a, NN, LL, HH, 1, 1, stream);
  gemm(t2, LL, Ws2b, LL, feat, LL, bs2b, NN, LL, LL, 0, 1, stream); // no relu

  // ---- gin_scorer conv1: s1 = relu(relu((M@feat) @ Wc1a + b) @ Wc1b + b) ----
  gemm(Mmat, NN, feat, LL, g3, LL, nullptr, NN, LL, NN, 0, 1, stream);
  gemm(g3, LL, Wc1a, HH, t3, HH, bc1a, NN, HH, LL, 1, 1, stream);
  gemm(t3, HH, Wc1b, HH, s1, HH, bc1b, NN, HH, HH, 1, 1, stream);   // relu fused

  // ---- gin_scorer conv2 (N=1) + heads ---------------------------------------
  gemm(Mmat, NN, s1, HH, g4, HH, nullptr, NN, HH, NN, 0, 1, stream);
  k_region<<<NN, 128, 0, stream>>>(g4, Wc2a, bc2a, Wc2b, bc2b, out);
  k_dementia<<<1, 256, 0, stream>>>(feat, Wd, bd, out);
}